// HGNNLayer_19868518711903
// MI455X (gfx1250) — compile-verified
//
#include <hip/hip_runtime.h>

typedef __attribute__((ext_vector_type(2))) float v2f;
typedef __attribute__((ext_vector_type(8))) float v8f;

#define D 128   // D_IN == D_OUT == 128

static __device__ __forceinline__ v8f wmma_f32(v2f a, v2f b, v8f c) {
  return __builtin_amdgcn_wmma_f32_16x16x4_f32(false, a, false, b, (short)0, c,
                                               false, false);
}

// ---------------------------------------------------------------------------
// Zero-fill scratch (degree counters) — ws is poisoned, re-zero every call.
// ---------------------------------------------------------------------------
__global__ void zero_f32(float* __restrict__ p, int n) {
  int i = blockIdx.x * blockDim.x + threadIdx.x;
  if (i < n) p[i] = 0.0f;
}

// ---------------------------------------------------------------------------
// Per-type in-degree: deg[dest] += 1 for each hyperedge. dest = ei1[e*S].
// ---------------------------------------------------------------------------
template <int S>
__global__ void degree_count(const long long* __restrict__ ei1,
                             float* __restrict__ deg, int nEdges) {
  int i = blockIdx.x * blockDim.x + threadIdx.x;
  if (i < nEdges) {
    long long d = ei1[(long long)i * S];
    atomicAdd(deg + d, 1.0f);
  }
}

// ---------------------------------------------------------------------------
// Stage weight matrix W[K][128] into LDS with K-pair interleave:
//   lds[(k>>1)*256 + n*2 + (k&1)] = W[k][n]
// so a B-fragment {W[k][n], W[k+1][n]} is ONE aligned ds_load_b64.
// ---------------------------------------------------------------------------
static __device__ __forceinline__ void stage_weights(float* lds,
                                                     const float* __restrict__ W,
                                                     int K) {
  for (int i = threadIdx.x; i < K * D; i += blockDim.x) {
    int k = i >> 7, n = i & (D - 1);
    lds[((k >> 1) << 8) + (n << 1) + (k & 1)] = W[i];
  }
  __syncthreads();
}

// ---------------------------------------------------------------------------
// Dense term: out = x @ C_w + C_b   (100000x128 @ 128x128), fp32 WMMA.
// One wave handles 32 rows x 128 cols (two 16x16x4 M-tiles sharing B frags).
// ---------------------------------------------------------------------------
__launch_bounds__(256)
__global__ void node_gemm_wmma(const float* __restrict__ x,
                               const float* __restrict__ Cw,
                               const float* __restrict__ Cb,
                               float* __restrict__ out, int nNodes) {
  extern __shared__ float ldsB[];  // 128x128 interleaved (64 KB)
  stage_weights(ldsB, Cw, D);

  const int lane = threadIdx.x & 31;
  const int wave = threadIdx.x >> 5;
  const int col  = lane & 15;  // N within 16-col strip; also A-matrix row m
  const int half = lane >> 4;  // ISA 16x4 f32 A layout: lanes16-31 hold K+2
  const int k2   = 2 * half;

  const int wpb = blockDim.x >> 5;
  const int nMT = (nNodes + 31) / 32;
  for (int mt = blockIdx.x * wpb + wave; mt < nMT; mt += gridDim.x * wpb) {
    const int r0 = mt * 32;
    int ra = r0 + col;       if (ra >= nNodes) ra = nNodes - 1;
    int rb = r0 + 16 + col;  if (rb >= nNodes) rb = nNodes - 1;
    const float* __restrict__ xa = x + (size_t)ra * D + k2;
    const float* __restrict__ xb = x + (size_t)rb * D + k2;

    v8f acc0[8] = {}, acc1[8] = {};
    v2f a0 = *(const v2f*)(xa);
    v2f a1 = *(const v2f*)(xb);
    for (int k0 = 0; k0 < D; k0 += 4) {
      const int kn = (k0 + 4 < D) ? k0 + 4 : 0;  // prefetch (dummy on last)
      v2f na0 = *(const v2f*)(xa + kn);
      v2f na1 = *(const v2f*)(xb + kn);
      const v2f* bp = (const v2f*)ldsB + (size_t)((k0 + k2) >> 1) * 128 + col;
#pragma unroll
      for (int n = 0; n < 8; ++n) {
        v2f b = bp[n * 16];
        acc0[n] = wmma_f32(a0, b, acc0[n]);
        acc1[n] = wmma_f32(a1, b, acc1[n]);
      }
      a0 = na0;
      a1 = na1;
    }

    // C/D layout: VGPR v, lanes0-15 -> row v, lanes16-31 -> row v+8.
#pragma unroll
    for (int v = 0; v < 8; ++v) {
      const int r = v + 8 * half;
      const int n0 = r0 + r;
      if (n0 < nNodes) {
        float* op = out + (size_t)n0 * D + col;
#pragma unroll
        for (int n = 0; n < 8; ++n) op[n * 16] = acc0[n][v] + Cb[n * 16 + col];
      }
      const int n1 = r0 + 16 + r;
      if (n1 < nNodes) {
        float* op = out + (size_t)n1 * D + col;
#pragma unroll
        for (int n = 0; n < 8; ++n) op[n * 16] = acc1[n][v] + Cb[n * 16 + col];
      }
    }
  }
}

// ---------------------------------------------------------------------------
// Edge aggregation for arity S: one wave per 32-edge macro-tile.
//   msg[32,128] = gather(x, src)[32, S*128] @ A[S*128, 128]   (fp32 WMMA)
//   out[dest]  += msg / deg[dest]                              (f32 atomics)
// A_r staged interleaved in LDS: 64/128/192 KB (fits 320 KB WGP LDS).
// ---------------------------------------------------------------------------
template <int S>
__launch_bounds__(256)
__global__ void edge_wmma(const float* __restrict__ x,
                          const long long* __restrict__ ei,  // [2, nE*S]
                          const float* __restrict__ A,       // [S*128, 128]
                          const float* __restrict__ deg,
                          float* __restrict__ out, int nEdges) {
  extern __shared__ float ldsB[];
  const int K = S * D;
  stage_weights(ldsB, A, K);

  const long long nTot = (long long)nEdges * S;
  const long long* __restrict__ ei0 = ei;         // sources
  const long long* __restrict__ ei1 = ei + nTot;  // dests

  const int lane = threadIdx.x & 31;
  const int wave = threadIdx.x >> 5;
  const int col  = lane & 15;
  const int half = lane >> 4;
  const int k2   = 2 * half;

  const int wpb = blockDim.x >> 5;
  const int nMT = (nEdges + 31) / 32;
  for (int mt = blockIdx.x * wpb + wave; mt < nMT; mt += gridDim.x * wpb) {
    const int e0 = mt * 32;
    int er0 = e0 + col;       if (er0 >= nEdges) er0 = nEdges - 1;
    int er1 = e0 + 16 + col;  if (er1 >= nEdges) er1 = nEdges - 1;

    const float* pa[S];  // per-slot gathered row base for tile 0 (this lane)
    const float* pb[S];  // ... tile 1
#pragma unroll
    for (int j = 0; j < S; ++j) {
      pa[j] = x + (size_t)ei0[(long long)er0 * S + j] * D + k2;
      pb[j] = x + (size_t)ei0[(long long)er1 * S + j] * D + k2;
    }

    v8f acc0[8] = {}, acc1[8] = {};
    v2f a0 = *(const v2f*)(pa[0]);
    v2f a1 = *(const v2f*)(pb[0]);
    for (int k0 = 0; k0 < K; k0 += 4) {
      const int kn   = (k0 + 4 < K) ? k0 + 4 : 0;
      const int slot = kn >> 7;
      const int kc   = kn & (D - 1);
      v2f na0 = *(const v2f*)(pa[slot] + kc);  // prefetch next K-step
      v2f na1 = *(const v2f*)(pb[slot] + kc);
      const v2f* bp = (const v2f*)ldsB + (size_t)((k0 + k2) >> 1) * 128 + col;
#pragma unroll
      for (int n = 0; n < 8; ++n) {
        v2f b = bp[n * 16];
        acc0[n] = wmma_f32(a0, b, acc0[n]);
        acc1[n] = wmma_f32(a1, b, acc1[n]);
      }
      a0 = na0;
      a1 = na1;
    }

    // Normalize by in-degree and scatter-add.
#pragma unroll
    for (int v = 0; v < 8; ++v) {
      const int r = v + 8 * half;
      const int ea = e0 + r;
      if (ea < nEdges) {
        const long long dst = ei1[(long long)ea * S];
        const float rdeg = 1.0f / deg[dst];
        float* op = out + (size_t)dst * D + col;
#pragma unroll
        for (int n = 0; n < 8; ++n) atomicAdd(op + n * 16, acc0[n][v] * rdeg);
      }
      const int eb = e0 + 16 + r;
      if (eb < nEdges) {
        const long long dst = ei1[(long long)eb * S];
        const float rdeg = 1.0f / deg[dst];
        float* op = out + (size_t)dst * D + col;
#pragma unroll
        for (int n = 0; n < 8; ++n) atomicAdd(op + n * 16, acc1[n][v] * rdeg);
      }
    }
  }
}

// ---------------------------------------------------------------------------
// Launch: inputs in setup_inputs() order:
//  0:x 1:A_r1 2:A_r2 3:A_r3 4:C_w 5:C_b 6:ei_r1 7:ei_r2 8:ei_r3 (int64)
// ---------------------------------------------------------------------------
extern "C" void kernel_launch(void* const* d_in, const int* in_sizes, int n_in,
                              void* d_out, int out_size, void* d_ws,
                              size_t ws_size, hipStream_t stream) {
  const float* x  = (const float*)d_in[0];
  const float* A1 = (const float*)d_in[1];
  const float* A2 = (const float*)d_in[2];
  const float* A3 = (const float*)d_in[3];
  const float* Cw = (const float*)d_in[4];
  const float* Cb = (const float*)d_in[5];
  const long long* e1 = (const long long*)d_in[6];
  const long long* e2 = (const long long*)d_in[7];
  const long long* e3 = (const long long*)d_in[8];

  const int nNodes = in_sizes[0] / D;
  const int nE1 = in_sizes[6] / 2;        // S=1
  const int nE2 = in_sizes[7] / (2 * 2);  // S=2
  const int nE3 = in_sizes[8] / (2 * 3);  // S=3

  float* deg1 = (float*)d_ws;
  float* deg2 = deg1 + nNodes;
  float* deg3 = deg2 + nNodes;

  // 1) zero degree counters
  {
    int n = 3 * nNodes;
    zero_f32<<<(n + 255) / 256, 256, 0, stream>>>(deg1, n);
  }

  // 2) in-degree per relation (dest row = second half of ei)
  degree_count<1><<<(nE1 + 255) / 256, 256, 0, stream>>>(
      e1 + (long long)nE1 * 1, deg1, nE1);
  degree_count<2><<<(nE2 + 255) / 256, 256, 0, stream>>>(
      e2 + (long long)nE2 * 2, deg2, nE2);
  degree_count<3><<<(nE3 + 255) / 256, 256, 0, stream>>>(
      e3 + (long long)nE3 * 3, deg3, nE3);

  // 3) dense term initializes out
  {
    int tiles = (nNodes + 31) / 32;
    int blocks = (tiles + 7) / 8;
    node_gemm_wmma<<<blocks, 256, (size_t)D * D * sizeof(float), stream>>>(
        x, Cw, Cb, (float*)d_out, nNodes);
  }

  // 4) edge aggregation (atomic adds into out)
  {
    int tiles = (nE1 + 31) / 32;
    int blocks = (tiles + 7) / 8;
    edge_wmma<1><<<blocks, 256, (size_t)1 * D * D * sizeof(float), stream>>>(
        x, e1, A1, deg1, (float*)d_out, nE1);
  }
  {
    int tiles = (nE2 + 31) / 32;
    int blocks = (tiles + 7) / 8;
    edge_wmma<2><<<blocks, 256, (size_t)2 * D * D * sizeof(float), stream>>>(
        x, e2, A2, deg2, (float*)d_out, nE2);
  }
  {
    int tiles = (nE3 + 31) / 32;
    int blocks = (tiles + 7) / 8;
    edge_wmma<3><<<blocks, 256, (size_t)3 * D * D * sizeof(float), stream>>>(
        x, e3, A3, deg3, (float*)d_out, nE3);
  }
}